// ReEig_28140625723419
// MI455X (gfx1250) — compile-verified
//
#include <hip/hip_runtime.h>

#define EPS_CLAMP 1e-4f
#define NN 64
#define ST 65          // LDS row stride (65 floats) -> conflict-free rows AND columns
#define NSWEEPS 10

typedef __attribute__((ext_vector_type(2))) float v2f;
typedef __attribute__((ext_vector_type(8))) float v8f;

__launch_bounds__(256)
__global__ void reeig_jacobi_wmma(const float* __restrict__ X, float* __restrict__ out) {
  __shared__ float As[NN * ST];   // working matrix, later W = V*diag(lam)
  __shared__ float Vs[NN * ST];   // accumulated eigenvectors
  __shared__ float lam[NN];
  __shared__ float cs[32], ss[32];
  __shared__ int   ps[32], qs[32];

  const int tid = threadIdx.x;
  const long long b = blockIdx.x;
  const float* __restrict__ Xb = X + b * (long long)(NN * NN);
  float* __restrict__ Ob = out + b * (long long)(NN * NN);

  // ---- load A (coalesced), init V = I ----
  for (int idx = tid; idx < NN * NN; idx += 256) {
    int i = idx >> 6, j = idx & 63;
    As[i * ST + j] = Xb[idx];
    Vs[i * ST + j] = (i == j) ? 1.0f : 0.0f;
  }
  __syncthreads();

  // ---- parallel cyclic Jacobi: 32 disjoint rotations per round ----
  for (int sweep = 0; sweep < NSWEEPS; ++sweep) {
    for (int r = 0; r < 63; ++r) {
      if (tid < 32) {
        int p, q;
        if (tid == 0) { p = 63; q = r; }
        else {
          p = r + tid;       if (p >= 63) p -= 63;
          q = r + 63 - tid;  if (q >= 63) q -= 63;
        }
        float app = As[p * ST + p];
        float aqq = As[q * ST + q];
        float apq = As[p * ST + q];
        float c = 1.0f, s = 0.0f;
        if (__builtin_fabsf(apq) > 1e-30f) {
          float tau = (aqq - app) / (2.0f * apq);
          float t = 1.0f / (__builtin_fabsf(tau) + __builtin_sqrtf(1.0f + tau * tau));
          if (tau < 0.0f) t = -t;
          c = 1.0f / __builtin_sqrtf(1.0f + t * t);
          s = t * c;
        }
        ps[tid] = p; qs[tid] = q; cs[tid] = c; ss[tid] = s;
      }
      __syncthreads();
      // rows: A <- J^T A   (pair k updates rows p,q; wave-contiguous columns)
      for (int idx = tid; idx < 2048; idx += 256) {
        int k = idx >> 6, j = idx & 63;
        int p = ps[k], q = qs[k];
        float c = cs[k], s = ss[k];
        float ap = As[p * ST + j], aq = As[q * ST + j];
        As[p * ST + j] = c * ap - s * aq;
        As[q * ST + j] = s * ap + c * aq;
      }
      __syncthreads();
      // cols: A <- A J,  V <- V J
      for (int idx = tid; idx < 2048; idx += 256) {
        int k = idx >> 6, i = idx & 63;
        int p = ps[k], q = qs[k];
        float c = cs[k], s = ss[k];
        float ap = As[i * ST + p], aq = As[i * ST + q];
        As[i * ST + p] = c * ap - s * aq;
        As[i * ST + q] = s * ap + c * aq;
        float vp = Vs[i * ST + p], vq = Vs[i * ST + q];
        Vs[i * ST + p] = c * vp - s * vq;
        Vs[i * ST + q] = s * vp + c * vq;
      }
      __syncthreads();
    }
  }

  // ---- ReEig clamp on (near-)diagonal, then W = V * diag(lam) into As ----
  if (tid < NN) lam[tid] = fmaxf(As[tid * ST + tid], EPS_CLAMP);
  __syncthreads();
  for (int idx = tid; idx < NN * NN; idx += 256) {
    int i = idx >> 6, j = idx & 63;
    As[i * ST + j] = Vs[i * ST + j] * lam[j];
  }
  __syncthreads();

  // ---- Out = W * V^T via V_WMMA_F32_16X16X4_F32 ----
  // 16 output tiles (4x4 grid of 16x16), 8 waves -> 2 tiles per wave.
  const int wave = tid >> 5;
  const int lane = tid & 31;
  const int half = lane >> 4;     // 0: lanes 0-15, 1: lanes 16-31
  const int l16  = lane & 15;

  for (int T = wave; T < 16; T += 8) {
    const int tm = T >> 2, tn = T & 3;
    v8f acc = {0.f, 0.f, 0.f, 0.f, 0.f, 0.f, 0.f, 0.f};
    const float* __restrict__ Wrow = &As[(tm * 16 + l16) * ST];  // A operand: row M=l16
    const float* __restrict__ Vrow = &Vs[(tn * 16 + l16) * ST];  // B operand: Vt[k][n] = V[n][k]
    #pragma unroll
    for (int k0 = 0; k0 < NN; k0 += 4) {
      const int k = k0 + 2 * half;          // ISA layout: vgpr0 holds K={0|2}, vgpr1 K={1|3}
      v2f a, bb;
      a.x  = Wrow[k];  a.y  = Wrow[k + 1];
      bb.x = Vrow[k];  bb.y = Vrow[k + 1];
      acc = __builtin_amdgcn_wmma_f32_16x16x4_f32(
          /*neg_a=*/false, a, /*neg_b=*/false, bb,
          /*c_mod=*/(short)0, acc, /*reuse_a=*/false, /*reuse_b=*/false);
    }
    // D layout: VGPR r -> M = r + 8*half, N = l16 (within tile)
    #pragma unroll
    for (int rr = 0; rr < 8; ++rr) {
      const int row = tm * 16 + rr + 8 * half;
      const int col = tn * 16 + l16;
      Ob[row * NN + col] = acc[rr];
    }
  }
}

extern "C" void kernel_launch(void* const* d_in, const int* in_sizes, int n_in,
                              void* d_out, int out_size, void* d_ws, size_t ws_size,
                              hipStream_t stream) {
  (void)n_in; (void)out_size; (void)d_ws; (void)ws_size;
  const float* X = (const float*)d_in[0];
  float* out = (float*)d_out;
  const int B = in_sizes[0] / (NN * NN);   // 8192
  reeig_jacobi_wmma<<<B, 256, 0, stream>>>(X, out);
}